// FKLoss_13846974562358
// MI455X (gfx1250) — compile-verified
//
#include <hip/hip_runtime.h>
#include <hip/hip_bf16.h>

typedef __attribute__((ext_vector_type(2))) float v2f;
typedef __attribute__((ext_vector_type(8))) float v8f;

#define TPB 256
#define NPER 4

// Wave-wide sum using V_WMMA_F32_16X16X4_F32.
// A holds one partial per lane (a.y = 0), B = all ones, C = 0.
// D[i][j] = sum_k A[i][k]  =>  any full column of D sums to the grand total of A.
// D layout (documented): lane l<16 holds rows M=0..7 of column N=l in its 8 VGPRs,
// lane l+16 holds rows M=8..15 of column N=l. So (sum of my 8 VGPRs) +
// (same from lane^16) == total over all 32 lanes' partials. EXEC must be all 1s.
__device__ __forceinline__ float wave_reduce_wmma(float s) {
  v2f a; a.x = s;    a.y = 0.0f;
  v2f b; b.x = 1.0f; b.y = 1.0f;
  v8f c = {};
  v8f d = __builtin_amdgcn_wmma_f32_16x16x4_f32(
      /*neg_a=*/false, a, /*neg_b=*/false, b,
      /*c_mod=*/(short)0, c, /*reuse_a=*/false, /*reuse_b=*/false);
  float p = 0.0f;
#pragma unroll
  for (int i = 0; i < 8; ++i) p += d[i];
  p += __shfl_xor(p, 16);
  return p;  // every lane holds the wave total
}

__global__ __launch_bounds__(TPB) void fk_partial_kernel(
    const float* __restrict__ joints, const float* __restrict__ poses,
    float* __restrict__ partial, int B) {
  // DH constants (compile-time; cos/sin of alpha fold to {0,1,-1})
  const float DD[7] = {0.333f, 0.0f, 0.316f, 0.0f, 0.384f, 0.0f, 0.107f};
  const float AA[7] = {0.0f, 0.0f, 0.0f, 0.0825f, -0.0825f, 0.0f, 0.088f};
  const float SA[7] = {0.0f, -1.0f, 1.0f, 1.0f, -1.0f, 1.0f, 1.0f};  // sin(alpha)
  const float CA[7] = {1.0f, 0.0f, 0.0f, 0.0f, 0.0f, 0.0f, 0.0f};    // cos(alpha)

  float s = 0.0f;
  const int base = blockIdx.x * (TPB * NPER) + threadIdx.x;

#pragma unroll
  for (int k = 0; k < NPER; ++k) {
    const int idx = base + k * TPB;
    if (idx < B) {
      const float* jp = joints + idx * 7;
      float R[3][3], P[3];
      // Joint 0 (alpha=0, a=0): T = Rz(theta) with d translation
      {
        float st, ct;
        sincosf(jp[0], &st, &ct);
        R[0][0] = ct;   R[0][1] = -st * CA[0]; R[0][2] = st * SA[0];
        R[1][0] = st;   R[1][1] = ct * CA[0];  R[1][2] = -ct * SA[0];
        R[2][0] = 0.0f; R[2][1] = SA[0];       R[2][2] = CA[0];
        P[0] = AA[0] * ct; P[1] = AA[0] * st; P[2] = DD[0];
      }
#pragma unroll
      for (int j = 1; j < 7; ++j) {
        float st, ct;
        sincosf(jp[j], &st, &ct);
        const float ca = CA[j], sa = SA[j], aa = AA[j], dd = DD[j];
        const float rj[3][3] = {{ct, -st * ca, st * sa},
                                {st, ct * ca, -ct * sa},
                                {0.0f, sa, ca}};
        const float pj[3] = {aa * ct, aa * st, dd};
        float Rn[3][3], Pn[3];
#pragma unroll
        for (int r = 0; r < 3; ++r) {
          Pn[r] = fmaf(R[r][0], pj[0], fmaf(R[r][1], pj[1], fmaf(R[r][2], pj[2], P[r])));
#pragma unroll
          for (int cc = 0; cc < 3; ++cc)
            Rn[r][cc] = fmaf(R[r][0], rj[0][cc],
                        fmaf(R[r][1], rj[1][cc], R[r][2] * rj[2][cc]));
        }
#pragma unroll
        for (int r = 0; r < 3; ++r) {
          P[r] = Pn[r];
#pragma unroll
          for (int cc = 0; cc < 3; ++cc) R[r][cc] = Rn[r][cc];
        }
      }
      // Euler XYZ from R
      const float e0 = atan2f(-R[1][2], R[2][2]);
      const float e1 = asinf(fminf(fmaxf(R[0][2], -1.0f), 1.0f));
      const float e2 = atan2f(-R[0][1], R[0][0]);
      const float* pp = poses + idx * 6;
      const float d0 = P[0] - pp[0], d1 = P[1] - pp[1], d2 = P[2] - pp[2];
      const float d3 = e0 - pp[3], d4 = e1 - pp[4], d5 = e2 - pp[5];
      s += d0 * d0 + d1 * d1 + d2 * d2 + d3 * d3 + d4 * d4 + d5 * d5;
    }
  }

  // WMMA intra-wave reduction (all lanes active here -> EXEC all 1s)
  const float wsum = wave_reduce_wmma(s);

  __shared__ float lsum[TPB / 32];
  const int lane = threadIdx.x & 31;
  const int wid = threadIdx.x >> 5;
  if (lane == 0) lsum[wid] = wsum;
  __syncthreads();
  if (threadIdx.x == 0) {
    float t = 0.0f;
#pragma unroll
    for (int w = 0; w < TPB / 32; ++w) t += lsum[w];
    partial[blockIdx.x] = t;
  }
}

__global__ __launch_bounds__(256) void fk_finish_kernel(
    const float* __restrict__ partial, int nparts, float inv,
    float* __restrict__ out) {
  __shared__ float sm[256];
  float t = 0.0f;
  for (int i = threadIdx.x; i < nparts; i += 256) t += partial[i];
  sm[threadIdx.x] = t;
  __syncthreads();
  for (int off = 128; off > 0; off >>= 1) {
    if ((int)threadIdx.x < off) sm[threadIdx.x] += sm[threadIdx.x + off];
    __syncthreads();
  }
  if (threadIdx.x == 0) out[0] = sm[0] * inv;
}

extern "C" void kernel_launch(void* const* d_in, const int* in_sizes, int n_in,
                              void* d_out, int out_size, void* d_ws, size_t ws_size,
                              hipStream_t stream) {
  const float* joints = (const float*)d_in[0];
  const float* poses = (const float*)d_in[1];
  float* out = (float*)d_out;
  float* partial = (float*)d_ws;

  const int B = in_sizes[0] / 7;
  const int blocks = (B + TPB * NPER - 1) / (TPB * NPER);  // 512 for B=524288
  fk_partial_kernel<<<blocks, TPB, 0, stream>>>(joints, poses, partial, B);

  const float inv = 1.0f / (6.0f * (float)B);
  fk_finish_kernel<<<1, 256, 0, stream>>>(partial, blocks, inv, out);
}